// MutationGCN_35175782154949
// MI455X (gfx1250) — compile-verified
//
#include <hip/hip_runtime.h>
#include <math.h>

#define N_NODES 50000
#define N_EDGES 1600000
#define IN_CH   256
#define HID_CH  128
#define NUM_CLS 64

typedef float v2f __attribute__((ext_vector_type(2)));
typedef float v8f __attribute__((ext_vector_type(8)));

// ---------------- degree / normalization ----------------

__global__ void init_deg_kernel(unsigned int* __restrict__ deg) {
  int i = blockIdx.x * blockDim.x + threadIdx.x;
  if (i < N_NODES) deg[i] = 1u;  // self-loop contributes 1 to every node's degree
}

__global__ void count_deg_kernel(const int* __restrict__ dst, unsigned int* __restrict__ deg) {
  int e = blockIdx.x * blockDim.x + threadIdx.x;
  if (e < N_EDGES) atomicAdd(&deg[dst[e]], 1u);
}

__global__ void dinv_kernel(const unsigned int* __restrict__ deg, float* __restrict__ dinv) {
  int i = blockIdx.x * blockDim.x + threadIdx.x;
  if (i < N_NODES) dinv[i] = rsqrtf((float)deg[i]);  // deg >= 1 always (self-loops)
}

// ---------------- f32 WMMA GEMM: C[M,NCOLS] = A[M,K] @ B[K,NCOLS] ----------------
// One block = one 16-row strip of A (M is an exact multiple of 16: 50000 = 3125*16).
// One wave per 16-column tile; K consumed 4 at a time via V_WMMA_F32_16X16X4_F32.
//
// A fragment (ISA 7.12.2, 32-bit A 16x4): lanes 0-15 -> M=lane, v0=K0,v1=K1;
//                                         lanes 16-31 -> M=lane-16, v0=K2,v1=K3.
// B fragment assumed as transposed-A layout: lane -> N, lane-group selects K pair.
// C/D (ISA table): VGPR r -> M=r (lanes 0-15) / M=8+r (lanes 16-31), N=lane%16.

template <int NCOLS, int K>
__global__ __launch_bounds__(32 * (NCOLS / 16))
void wmma_gemm_f32(const float* __restrict__ A, const float* __restrict__ B,
                   float* __restrict__ C) {
  __shared__ float As[16][K + 4];  // +4 pad: LDS column reads hit banks 0,4,...,60
  const int m0  = blockIdx.x * 16;
  const int tid = threadIdx.x;
  constexpr int NT = 32 * (NCOLS / 16);
  constexpr int V  = 16 * K / 4;  // float4 chunks in the A strip

  for (int i = tid; i < V; i += NT) {
    int r = i / (K / 4);
    int c = i % (K / 4);
    float4 v = ((const float4*)(A + (size_t)(m0 + r) * K))[c];
    *(float4*)&As[r][c * 4] = v;
  }
  __syncthreads();

  const int lane = tid & 31;
  const int n0   = (tid >> 5) * 16;
  const int lr   = lane & 15;          // M (for A) / N (for B,C)
  const int kh   = (lane >> 4) * 2;    // K pair selector: 0 or 2

  v8f acc = {};
  for (int k = 0; k < K; k += 4) {
    v2f a, b;
    a.x = As[lr][k + kh];
    a.y = As[lr][k + kh + 1];
    b.x = B[(size_t)(k + kh) * NCOLS + n0 + lr];
    b.y = B[(size_t)(k + kh + 1) * NCOLS + n0 + lr];
    acc = __builtin_amdgcn_wmma_f32_16x16x4_f32(
        /*neg_a=*/false, a, /*neg_b=*/false, b,
        /*c_mod=*/(short)0, acc, /*reuse_a=*/false, /*reuse_b=*/false);
  }

  const int rowbase = m0 + ((lane >> 4) << 3);  // +8 for upper lane half
#pragma unroll
  for (int r = 0; r < 8; ++r)
    C[(size_t)(rowbase + r) * NCOLS + n0 + lr] = acc[r];
}

// ---------------- message passing ----------------

// agg[i,:] = dinv[i]^2 * h[i,:]   (self-loop term; also initializes the accumulator)
template <int C>
__global__ void selfloop_init_kernel(const float* __restrict__ h,
                                     const float* __restrict__ dinv,
                                     float* __restrict__ agg) {
  int t = blockIdx.x * blockDim.x + threadIdx.x;
  if (t >= N_NODES * C) return;
  float di = dinv[t / C];
  agg[t] = di * di * h[t];
}

// One wave per edge; lane handles C/32 consecutive floats (float4 or float2).
template <int C>
__global__ void scatter_edges_kernel(const int* __restrict__ src, const int* __restrict__ dst,
                                     const float* __restrict__ h, const float* __restrict__ dinv,
                                     float* __restrict__ agg) {
  int t = blockIdx.x * blockDim.x + threadIdx.x;
  int e = t >> 5;
  if (e >= N_EDGES) return;
  int lane = t & 31;
  int s = src[e], d = dst[e];
  float nrm = dinv[s] * dinv[d];
  constexpr int PER = C / 32;
  const float* hp = h + (size_t)s * C + lane * PER;
  float* ap = agg + (size_t)d * C + lane * PER;
  if constexpr (PER == 4) {
    float4 v = *(const float4*)hp;
    atomicAdd(ap + 0, v.x * nrm);
    atomicAdd(ap + 1, v.y * nrm);
    atomicAdd(ap + 2, v.z * nrm);
    atomicAdd(ap + 3, v.w * nrm);
  } else {
    float2 v = *(const float2*)hp;
    atomicAdd(ap + 0, v.x * nrm);
    atomicAdd(ap + 1, v.y * nrm);
  }
}

// ---------------- epilogues ----------------

__global__ void bias_relu_kernel(float* __restrict__ a, const float* __restrict__ b) {
  int t = blockIdx.x * blockDim.x + threadIdx.x;
  if (t >= N_NODES * HID_CH) return;
  a[t] = fmaxf(a[t] + b[t & (HID_CH - 1)], 0.0f);
}

__global__ void output_kernel(const float* __restrict__ agg, const float* __restrict__ b,
                              float* __restrict__ out) {
  int t = blockIdx.x * blockDim.x + threadIdx.x;
  if (t >= N_NODES * NUM_CLS) return;
  float z = agg[t] + b[t & (NUM_CLS - 1)];
  float p = 1.0f / (1.0f + expf(-z));
  out[t] = (p > 0.5f) ? 1.0f : 0.0f;          // predictions
  out[(size_t)N_NODES * NUM_CLS + t] = p;     // probabilities
}

// ---------------- launch ----------------

extern "C" void kernel_launch(void* const* d_in, const int* in_sizes, int n_in,
                              void* d_out, int out_size, void* d_ws, size_t ws_size,
                              hipStream_t stream) {
  (void)in_sizes; (void)n_in; (void)out_size; (void)ws_size;
  const float* x  = (const float*)d_in[0];
  const int*   ei = (const int*)d_in[1];
  const float* W1 = (const float*)d_in[2];
  const float* b1 = (const float*)d_in[3];
  const float* W2 = (const float*)d_in[4];
  const float* b2 = (const float*)d_in[5];
  float* out = (float*)d_out;

  const int* srcE = ei;             // edge_index[0]
  const int* dstE = ei + N_EDGES;   // edge_index[1]

  char* ws = (char*)d_ws;
  size_t off = 0;
  auto take = [&](size_t bytes) {
    char* p = ws + off;
    off = (off + bytes + 255) & ~(size_t)255;
    return p;
  };
  unsigned int* deg  = (unsigned int*)take((size_t)N_NODES * 4);
  float*        dinv = (float*)take((size_t)N_NODES * 4);
  float*        h1   = (float*)take((size_t)N_NODES * HID_CH * 4);
  float*        agg1 = (float*)take((size_t)N_NODES * HID_CH * 4);
  float*        h2   = (float*)take((size_t)N_NODES * NUM_CLS * 4);
  float*        agg2 = (float*)take((size_t)N_NODES * NUM_CLS * 4);

  const dim3 blk(256);
  init_deg_kernel<<<(N_NODES + 255) / 256, blk, 0, stream>>>(deg);
  count_deg_kernel<<<(N_EDGES + 255) / 256, blk, 0, stream>>>(dstE, deg);
  dinv_kernel<<<(N_NODES + 255) / 256, blk, 0, stream>>>(deg, dinv);

  // Layer 1
  wmma_gemm_f32<HID_CH, IN_CH><<<N_NODES / 16, 32 * (HID_CH / 16), 0, stream>>>(x, W1, h1);
  selfloop_init_kernel<HID_CH>
      <<<((size_t)N_NODES * HID_CH + 255) / 256, blk, 0, stream>>>(h1, dinv, agg1);
  scatter_edges_kernel<HID_CH>
      <<<((size_t)N_EDGES * 32 + 255) / 256, blk, 0, stream>>>(srcE, dstE, h1, dinv, agg1);
  bias_relu_kernel<<<((size_t)N_NODES * HID_CH + 255) / 256, blk, 0, stream>>>(agg1, b1);

  // Layer 2
  wmma_gemm_f32<NUM_CLS, HID_CH><<<N_NODES / 16, 32 * (NUM_CLS / 16), 0, stream>>>(agg1, W2, h2);
  selfloop_init_kernel<NUM_CLS>
      <<<((size_t)N_NODES * NUM_CLS + 255) / 256, blk, 0, stream>>>(h2, dinv, agg2);
  scatter_edges_kernel<NUM_CLS>
      <<<((size_t)N_EDGES * 32 + 255) / 256, blk, 0, stream>>>(srcE, dstE, h2, dinv, agg2);
  output_kernel<<<((size_t)N_NODES * NUM_CLS + 255) / 256, blk, 0, stream>>>(agg2, b2, out);
}